// GatAggregateFlatten_7009386627256
// MI455X (gfx1250) — compile-verified
//
#include <hip/hip_runtime.h>
#include <hip/hip_bf16.h>
#include <math.h>

// ---------------------------------------------------------------------------
// GAT aggregate + flatten, algebraically reduced:
//   ws   = W @ a_src                      (128)
//   s_bn = leakyrelu(x_bn . ws)           (scores; a_dst term is exactly 0)
//   xbar_b = sum_n softmax(s)_n * x_bn    (online softmax, single pass over x)
//   y_b  = LayerNorm(xbar_b @ W) * gamma + beta   (WMMA f32 16x16x4)
// ---------------------------------------------------------------------------

typedef __attribute__((ext_vector_type(2))) float v2f;
typedef __attribute__((ext_vector_type(8))) float v8f;

#define B_ 32
#define N_ 8192
#define F_ 128
#define CHUNKS 32
#define NODES_PER_BLOCK (N_ / CHUNKS)              // 256
#define WAVES_PER_BLOCK 8
#define NODES_PER_WAVE (NODES_PER_BLOCK / WAVES_PER_BLOCK) // 32
#define NEG_SLOPE 0.2f
#define LN_EPS 1e-5f
#define REC_STRIDE 132   // floats per partial record: m, l, pad2, acc[128]
#define NEG_BIG (-3.402823e38f)

// ---- kernel 1: ws[f] = sum_o W[f][o] * a_src[o] ---------------------------
__global__ void k_ws(const float* __restrict__ W, const float* __restrict__ a_src,
                     float* __restrict__ wsv) {
    int f = threadIdx.x;          // 128 threads
    float acc = 0.f;
    for (int o = 0; o < F_; ++o) acc = fmaf(W[f * F_ + o], a_src[o], acc);
    wsv[f] = acc;
}

// ---- kernel 2: streaming pass, per-wave online softmax --------------------
__global__ void __launch_bounds__(256)
k_scan(const float* __restrict__ x, const unsigned char* __restrict__ mask,
       const float* __restrict__ wsv, float* __restrict__ partials) {
    const int chunk = blockIdx.x;
    const int b     = blockIdx.y;
    const int lane  = threadIdx.x & 31;
    const int w     = threadIdx.x >> 5;

    const float4 ws4 = ((const float4*)wsv)[lane];

    float  m = NEG_BIG, l = 0.f;
    float4 acc = make_float4(0.f, 0.f, 0.f, 0.f);

    const int n0 = chunk * NODES_PER_BLOCK + w * NODES_PER_WAVE;
    const float4* xb = (const float4*)(x + (size_t)b * N_ * F_);
    const unsigned char* mb = mask + (size_t)b * N_;

    for (int i = 0; i < NODES_PER_WAVE; ++i) {
        const int n = n0 + i;
        if (i + 4 < NODES_PER_WAVE)
            __builtin_prefetch(&xb[(size_t)(n + 4) * 32 + lane], 0, 0);
        const float4 x4 = xb[(size_t)n * 32 + lane];

        // per-lane partial dot, then wave32 all-reduce
        float p = x4.x * ws4.x + x4.y * ws4.y + x4.z * ws4.z + x4.w * ws4.w;
        #pragma unroll
        for (int off = 16; off; off >>= 1) p += __shfl_xor(p, off, 32);

        float s = (p > 0.f) ? p : NEG_SLOPE * p;   // leaky relu, T=1
        if (mb[n]) {                               // uniform across the wave
            const float mn    = fmaxf(m, s);
            const float scale = __expf(m - mn);
            const float pe    = __expf(s - mn);
            l = l * scale + pe;
            acc.x = acc.x * scale + pe * x4.x;
            acc.y = acc.y * scale + pe * x4.y;
            acc.z = acc.z * scale + pe * x4.z;
            acc.w = acc.w * scale + pe * x4.w;
            m = mn;
        }
    }

    // combine the 8 wave states inside the block
    __shared__ float sm[WAVES_PER_BLOCK], sl[WAVES_PER_BLOCK];
    __shared__ float sacc[WAVES_PER_BLOCK][F_];
    if (lane == 0) { sm[w] = m; sl[w] = l; }
    ((float4*)sacc[w])[lane] = acc;
    __syncthreads();

    if (threadIdx.x < F_) {
        const int t = threadIdx.x;
        float M = NEG_BIG, L = 0.f, A = 0.f;
        for (int j = 0; j < WAVES_PER_BLOCK; ++j) {
            const float mj = sm[j], lj = sl[j];
            const float Mn = fmaxf(M, mj);
            const float s0 = __expf(M - Mn), s1 = __expf(mj - Mn);
            L = L * s0 + lj * s1;
            A = A * s0 + sacc[j][t] * s1;
            M = Mn;
        }
        float* rec = partials + ((size_t)b * CHUNKS + chunk) * REC_STRIDE;
        if (t == 0) { rec[0] = M; rec[1] = L; }
        rec[4 + t] = A;
    }
}

// ---- kernel 3: reduce chunk partials -> xbar[b][f] ------------------------
__global__ void k_reduce(const float* __restrict__ partials, float* __restrict__ xbar) {
    const int b = blockIdx.x;
    const int t = threadIdx.x;    // 128 threads
    float M = NEG_BIG, L = 0.f, A = 0.f;
    for (int c = 0; c < CHUNKS; ++c) {
        const float* rec = partials + ((size_t)b * CHUNKS + c) * REC_STRIDE;
        const float mc = rec[0], lc = rec[1], ac = rec[4 + t];
        const float Mn = fmaxf(M, mc);
        const float s0 = __expf(M - Mn), s1 = __expf(mc - Mn);
        L = L * s0 + lc * s1;
        A = A * s0 + ac * s1;
        M = Mn;
    }
    xbar[b * F_ + t] = A / L;
}

// ---- kernel 4: H = xbar @ W via V_WMMA_F32_16X16X4_F32, then LayerNorm ----
__global__ void __launch_bounds__(512)
k_gemm_ln(const float* __restrict__ xbar, const float* __restrict__ W,
          const float* __restrict__ gamma, const float* __restrict__ beta,
          float* __restrict__ out) {
    __shared__ float H[B_][F_];

    const int lane = threadIdx.x & 31;
    const int wv   = threadIdx.x >> 5;  // 16 waves
    const int mt   = wv >> 3;           // 0..1  (M tile)
    const int nt   = wv & 7;            // 0..7  (N tile)
    const int half = lane >> 4;         // 0 -> K=0,1 ; 1 -> K=2,3
    const int l15  = lane & 15;

    v8f c = {};
    for (int k = 0; k < F_; k += 4) {
        const int ka = k + 2 * half;
        // A 16x4 f32 fragment: lanes 0-15 hold M=l15 K=0,1; lanes 16-31 K=2,3
        const int arow = mt * 16 + l15;
        v2f a;
        a.x = xbar[arow * F_ + ka + 0];
        a.y = xbar[arow * F_ + ka + 1];
        // B 4x16 f32 fragment: lanes 0-15 hold N=l15 K=0,1; lanes 16-31 K=2,3
        const int bcol = nt * 16 + l15;
        v2f bf;
        bf.x = W[(ka + 0) * F_ + bcol];
        bf.y = W[(ka + 1) * F_ + bcol];
        c = __builtin_amdgcn_wmma_f32_16x16x4_f32(false, a, false, bf,
                                                  (short)0, c, false, false);
    }

    // scatter 16x16 f32 tile: VGPR i -> rows i (lanes 0-15) / i+8 (lanes 16-31)
    #pragma unroll
    for (int i = 0; i < 8; ++i) {
        const int row = mt * 16 + i + 8 * half;
        const int col = nt * 16 + l15;
        H[row][col] = c[i];
    }
    __syncthreads();

    // LayerNorm: wave wv handles rows 2*wv and 2*wv+1
    #pragma unroll
    for (int rr = 0; rr < 2; ++rr) {
        const int row = wv * 2 + rr;
        const float4 h4 = ((const float4*)H[row])[lane];
        float s = h4.x + h4.y + h4.z + h4.w;
        float q = h4.x * h4.x + h4.y * h4.y + h4.z * h4.z + h4.w * h4.w;
        #pragma unroll
        for (int off = 16; off; off >>= 1) {
            s += __shfl_xor(s, off, 32);
            q += __shfl_xor(q, off, 32);
        }
        const float mu  = s * (1.f / F_);
        const float var = q * (1.f / F_) - mu * mu;
        const float inv = rsqrtf(var + LN_EPS);
        const float4 g4 = ((const float4*)gamma)[lane];
        const float4 b4 = ((const float4*)beta)[lane];
        float4 y;
        y.x = (h4.x - mu) * inv * g4.x + b4.x;
        y.y = (h4.y - mu) * inv * g4.y + b4.y;
        y.z = (h4.z - mu) * inv * g4.z + b4.z;
        y.w = (h4.w - mu) * inv * g4.w + b4.w;
        ((float4*)(out + row * F_))[lane] = y;
    }
}

// ---------------------------------------------------------------------------
extern "C" void kernel_launch(void* const* d_in, const int* in_sizes, int n_in,
                              void* d_out, int out_size, void* d_ws, size_t ws_size,
                              hipStream_t stream) {
    const float*         x     = (const float*)d_in[0];
    const unsigned char* mask  = (const unsigned char*)d_in[1]; // bool array
    const float*         W     = (const float*)d_in[2];
    const float*         a_src = (const float*)d_in[3];
    // d_in[4] = a_dst: provably contributes 0 (root row of x is zeros)
    const float*         gamma = (const float*)d_in[5];
    const float*         beta  = (const float*)d_in[6];
    float*               out   = (float*)d_out;

    float* wsf      = (float*)d_ws;
    float* wsv      = wsf;                       // 128 floats
    float* partials = wsf + 128;                 // B*CHUNKS*REC_STRIDE floats
    float* xbar     = partials + (size_t)B_ * CHUNKS * REC_STRIDE; // B*F floats

    k_ws<<<1, 128, 0, stream>>>(W, a_src, wsv);

    dim3 g2(CHUNKS, B_);
    k_scan<<<g2, 256, 0, stream>>>(x, mask, wsv, partials);

    k_reduce<<<B_, 128, 0, stream>>>(partials, xbar);

    k_gemm_ln<<<1, 512, 0, stream>>>(xbar, W, gamma, beta, out);
}